// Dsa_Decoder_43722767073505
// MI455X (gfx1250) — compile-verified
//
#include <hip/hip_runtime.h>
#include <math.h>

// ---------------------------------------------------------------------------
// Dsa_Decoder collapsed implementation for MI455X (gfx1250, wave32).
//
// Exact-math collapse of the reference:
//   * log_softmax over a singleton axis == 0.0 exactly  -> beta == 0
//   * => ctx carry == 0 at every step, enc_h/w1/w2/v are dead
//   * lstm_cell uses fixed (h0, c0)   -> only the LAST scan step matters
//   * surviving work: gates = h0 @ w_hh^T + b_ih + b_hh + x*w_ih  (512x256x64
//     fp32 GEMM) followed by the elementwise LSTM combine.
// GEMM is done with the native fp32 WMMA: V_WMMA_F32_16X16X4_F32.
// ---------------------------------------------------------------------------

#define BB 512   // batch
#define HH 64    // hidden
#define TSTEPS 64

typedef float v2f __attribute__((ext_vector_type(2)));
typedef float v8f __attribute__((ext_vector_type(8)));

// Kernel 1: gates(B,4H) = h_init(B,H) @ w_hh^T(H,4H) + (b_ih + b_hh)
// One wave computes one 16x16 tile of the 512x256 result via 16 chained
// v_wmma_f32_16x16x4_f32 ops (K=64 in steps of 4).
__global__ __launch_bounds__(128) void dsa_gates_wmma(
    const float* __restrict__ h0,     // (B,H) row-major
    const float* __restrict__ w_hh,   // (4H,H) row-major; B[k][n] = w_hh[n][k]
    const float* __restrict__ b_ih,   // (4H)
    const float* __restrict__ b_hh,   // (4H)
    float* __restrict__ gates)        // (B,4H)
{
    const int lane = threadIdx.x & 31;
    const int wv   = threadIdx.x >> 5;
    const int tile = blockIdx.x * 4 + wv;   // 0..511
    const int tM   = tile >> 4;             // 0..31 : 16-row batch tile
    const int tN   = tile & 15;             // 0..15 : 16-col gate tile

    const int l15 = lane & 15;
    const int kp  = (lane >> 4) << 1;       // lanes 0-15 -> K+{0,1}; 16-31 -> K+{2,3}

    const int arow = tM * 16 + l15;         // batch row held by this lane (A frag)
    const int bcol = tN * 16 + l15;         // gate column held by this lane (B frag)

    const float* aptr = h0   + arow * HH;   // A[m][k], contiguous in k
    const float* bptr = w_hh + bcol * HH;   // B[k][n] = w_hh[n][k], contiguous in k

    v8f acc = {0.f, 0.f, 0.f, 0.f, 0.f, 0.f, 0.f, 0.f};
#pragma unroll
    for (int k0 = 0; k0 < HH; k0 += 4) {
        v2f a = *(const v2f*)(aptr + k0 + kp);   // 8B-aligned float2 loads
        v2f b = *(const v2f*)(bptr + k0 + kp);
        acc = __builtin_amdgcn_wmma_f32_16x16x4_f32(
                  /*neg_a=*/false, a, /*neg_b=*/false, b,
                  /*c_mod=*/(short)0, acc, /*reuse_a=*/false, /*reuse_b=*/false);
    }

    // C/D layout: VGPR r -> M=r (lanes 0-15) or M=8+r (lanes 16-31), N=lane&15.
    const float bias  = b_ih[bcol] + b_hh[bcol];
    const int   mbase = tM * 16 + ((lane >> 4) << 3);
#pragma unroll
    for (int r = 0; r < 8; ++r) {
        gates[(mbase + r) * 256 + bcol] = acc[r] + bias;
    }
}

// Kernel 2: per-(b,h) LSTM combine for the final step only.
//   x[b]  = t[b, T-2] * dense_w[0,0] + dense_b[0]        (ctx contribution == 0)
//   c2    = sig(f)*c0 + sig(i)*tanh(g);  h2 = sig(o)*tanh(c2)
//   out[b, 0:64] = h2 ; out[b, 64:128] = ctx == 0
__global__ __launch_bounds__(256) void dsa_finish(
    const float* __restrict__ t,        // (B, T-1)
    const float* __restrict__ c0,       // (B,H)
    const float* __restrict__ dense_w,  // (1, H+1)
    const float* __restrict__ dense_b,  // (1)
    const float* __restrict__ w_ih,     // (4H,1)
    const float* __restrict__ gates,    // (B,4H)
    float* __restrict__ out)            // (B, 2H)
{
    const int idx = blockIdx.x * blockDim.x + threadIdx.x;  // 0..B*H-1
    const int b = idx >> 6;
    const int h = idx & 63;

    const float xb = t[b * (TSTEPS - 1) + (TSTEPS - 2)] * dense_w[0] + dense_b[0];

    const float gi = gates[b * 256 +       h] + xb * w_ih[      h];
    const float gf = gates[b * 256 +  64 + h] + xb * w_ih[ 64 + h];
    const float gg = gates[b * 256 + 128 + h] + xb * w_ih[128 + h];
    const float go = gates[b * 256 + 192 + h] + xb * w_ih[192 + h];

    const float cin = c0[b * HH + h];
    const float si  = 1.0f / (1.0f + expf(-gi));
    const float sf  = 1.0f / (1.0f + expf(-gf));
    const float so  = 1.0f / (1.0f + expf(-go));

    const float c2 = sf * cin + si * tanhf(gg);
    const float h2 = so * tanhf(c2);

    out[b * 128 +      h] = h2;
    out[b * 128 + 64 + h] = 0.0f;   // ctx is exactly zero
}

extern "C" void kernel_launch(void* const* d_in, const int* in_sizes, int n_in,
                              void* d_out, int out_size, void* d_ws, size_t ws_size,
                              hipStream_t stream) {
    // setup_inputs() order (all fp32):
    const float* t    = (const float*)d_in[0];   // (512,63)
    // d_in[1] = enc_h : dead (beta == 0 exactly)
    const float* h0   = (const float*)d_in[2];   // (1,512,64) -> (512,64)
    const float* c0   = (const float*)d_in[3];   // (1,512,64) -> (512,64)
    const float* dw   = (const float*)d_in[4];   // dense_w (1,65)
    const float* db   = (const float*)d_in[5];   // dense_b (1)
    const float* wih  = (const float*)d_in[6];   // w_ih (256,1)
    const float* whh  = (const float*)d_in[7];   // w_hh (256,64)
    const float* bih  = (const float*)d_in[8];   // b_ih (256)
    const float* bhh  = (const float*)d_in[9];   // b_hh (256)
    // d_in[10..15] = w1_*, w2_*, v_* : dead

    float* gates = (float*)d_ws;                 // 512*256 fp32 = 512 KB scratch

    // 512 tiles, 4 waves/block -> 128 blocks of 128 threads (all lanes active).
    dsa_gates_wmma<<<128, 128, 0, stream>>>(h0, whh, bih, bhh, gates);
    // 512*64 threads for the elementwise LSTM combine + output assembly.
    dsa_finish<<<128, 256, 0, stream>>>(t, c0, dw, db, wih, gates, (float*)d_out);
}